// DistributionalVariancePenalization_57045755625727
// MI455X (gfx1250) — compile-verified
//
#include <hip/hip_runtime.h>
#include <math.h>

// ---------------- configuration ----------------
#define NT      256            // threads per block (8 wave32 waves)
#define NCHUNK  8              // float4 chunks per thread
#define N_ITER  48             // scheduled root-find iterations (early-exit once converged)
#define DVP_TOL 1e-4
#define MIN_REL_DIFFERENCE 1e-5f

typedef float f32x4 __attribute__((ext_vector_type(4)));

struct DvpState {
    float vmax, vmin, size, meanv;
    float emin, emax, eta, eta_star;
    int   mode;   // 0 = bisection result, 1 = flat -> mean, 2 = small -> vmax
    int   done;
    int   iter;
    int   pad;
};

// ---------------- CDNA5 async global->LDS path ----------------
#if defined(__has_builtin)
#  if __has_builtin(__builtin_amdgcn_global_load_async_to_lds_b128)
#    define HAS_ASYNC_BUILTIN 1
#  else
#    define HAS_ASYNC_BUILTIN 0
#  endif
#  if __has_builtin(__builtin_amdgcn_s_wait_asynccnt)
#    define HAS_WAIT_BUILTIN 1
#  else
#    define HAS_WAIT_BUILTIN 0
#  endif
#else
#  define HAS_ASYNC_BUILTIN 0
#  define HAS_WAIT_BUILTIN 0
#endif

#define COMPILER_BARRIER() asm volatile("" ::: "memory")

// Match the builtin's exact parameter types:
//   arg0: int __attribute__((vector_size(16))) * in AS1 (global), non-const
//   arg1: same pointee in AS3 (LDS)
typedef int v4i_vs __attribute__((vector_size(16)));
typedef __attribute__((address_space(1))) v4i_vs* g_v4i_ptr;
typedef __attribute__((address_space(3))) v4i_vs* l_v4i_ptr;

__device__ __forceinline__ void asyncCopy16(const f32x4* g, f32x4* lds) {
#if HAS_ASYNC_BUILTIN
    __builtin_amdgcn_global_load_async_to_lds_b128(
        (g_v4i_ptr)(v4i_vs*)g,
        (l_v4i_ptr)(v4i_vs*)lds,
        0, 0);
#else
    // VDST = LDS byte address (low 32 bits of generic shared pointer per
    // aperture rules), VADDR = 64-bit global address, no SADDR.
    unsigned lds32 = (unsigned)(unsigned long long)lds;
    asm volatile("global_load_async_to_lds_b128 %0, %1, off"
                 :: "v"(lds32), "v"(g) : "memory");
#endif
}

__device__ __forceinline__ void waitAsync0() {
#if HAS_WAIT_BUILTIN
    __builtin_amdgcn_s_wait_asynccnt(0);
#else
    asm volatile("s_wait_asynccnt 0" ::: "memory");
#endif
    COMPILER_BARRIER();
}
__device__ __forceinline__ void waitAsync1() {
#if HAS_WAIT_BUILTIN
    __builtin_amdgcn_s_wait_asynccnt(1);
#else
    asm volatile("s_wait_asynccnt 1" ::: "memory");
#endif
    COMPILER_BARRIER();
}

// ---------------- block reductions ----------------
__device__ __forceinline__ float blockSumF(float v, float* sm) {
    int t = threadIdx.x;
    __syncthreads(); sm[t] = v; __syncthreads();
    for (int s = NT / 2; s > 0; s >>= 1) { if (t < s) sm[t] += sm[t + s]; __syncthreads(); }
    return sm[0];
}
__device__ __forceinline__ float blockMaxF(float v, float* sm) {
    int t = threadIdx.x;
    __syncthreads(); sm[t] = v; __syncthreads();
    for (int s = NT / 2; s > 0; s >>= 1) { if (t < s) sm[t] = fmaxf(sm[t], sm[t + s]); __syncthreads(); }
    return sm[0];
}
__device__ __forceinline__ float blockMinF(float v, float* sm) {
    int t = threadIdx.x;
    __syncthreads(); sm[t] = v; __syncthreads();
    for (int s = NT / 2; s > 0; s >>= 1) { if (t < s) sm[t] = fminf(sm[t], sm[t + s]); __syncthreads(); }
    return sm[0];
}
__device__ __forceinline__ double blockSumD(double v, double* sm) {
    int t = threadIdx.x;
    __syncthreads(); sm[t] = v; __syncthreads();
    for (int s = NT / 2; s > 0; s >>= 1) { if (t < s) sm[t] += sm[t + s]; __syncthreads(); }
    return sm[0];
}

// ---------------- pass 1: stats (sum, sumsq, max, min) ----------------
__global__ void dvp_stats_partial(const f32x4* __restrict__ v4,
                                  float* __restrict__ part, int n4) {
    __shared__ f32x4 buf[2][NT];
    __shared__ float sm[NT];
    const int tid = threadIdx.x;
    const long long base = (long long)blockIdx.x * (NT * NCHUNK) + tid;

    float s = 0.f, s2 = 0.f;
    float mx = -__builtin_inff(), mn = __builtin_inff();

    if (base < n4) asyncCopy16(v4 + base, &buf[0][tid]);
    COMPILER_BARRIER();
    for (int c = 0; c < NCHUNK; ++c) {
        long long idx = base + (long long)c * NT;
        if (c + 1 < NCHUNK) {
            long long nidx = base + (long long)(c + 1) * NT;
            if (nidx < n4) asyncCopy16(v4 + nidx, &buf[(c + 1) & 1][tid]);
            COMPILER_BARRIER();
            waitAsync1();
        } else {
            waitAsync0();
        }
        f32x4 x = buf[c & 1][tid];
        if (idx < n4) {
            s += x.x + x.y + x.z + x.w;
            s2 = fmaf(x.x, x.x, fmaf(x.y, x.y, fmaf(x.z, x.z, fmaf(x.w, x.w, s2))));
            mx = fmaxf(mx, fmaxf(fmaxf(x.x, x.y), fmaxf(x.z, x.w)));
            mn = fminf(mn, fminf(fminf(x.x, x.y), fminf(x.z, x.w)));
        }
    }
    float bs  = blockSumF(s, sm);
    float bs2 = blockSumF(s2, sm);
    float bmx = blockMaxF(mx, sm);
    float bmn = blockMinF(mn, sm);
    if (tid == 0) {
        float* p = part + (size_t)blockIdx.x * 8;
        p[0] = bs; p[1] = bs2; p[2] = bmx; p[3] = bmn;
    }
}

__global__ void dvp_stats_final(const float* __restrict__ part,
                                DvpState* __restrict__ st, int nb, double dm) {
    __shared__ double dsm[NT];
    __shared__ float  fsm[NT];
    const int t = threadIdx.x;
    double s = 0.0, s2 = 0.0;
    float mx = -__builtin_inff(), mn = __builtin_inff();
    for (int i = t; i < nb; i += NT) {
        const float* p = part + (size_t)i * 8;
        s += (double)p[0]; s2 += (double)p[1];
        mx = fmaxf(mx, p[2]); mn = fminf(mn, p[3]);
    }
    s  = blockSumD(s,  dsm);
    s2 = blockSumD(s2, dsm);
    mx = blockMaxF(mx, fsm);
    mn = blockMinF(mn, fsm);
    if (t == 0) {
        double mean = s / dm;
        double var  = (s2 - s * s / dm) / (dm - 1.0);   // unbiased (ddof=1)
        float size  = (float)var;                        // LMBDA=1, GAMMA=1 -> var^1
        float vmax = mx, vmin = mn;
        float eta_max = vmax;
        float eta_min = -(1.0f / (sqrtf(2.0f * size + 1.0f) - 1.0f)) * vmax;
        int mode = 0;
        if ((vmax - vmin) / vmax <= MIN_REL_DIFFERENCE) mode = 1;        // flat -> uniform q
        else if (dm <= 1.0 + 2.0 * (double)size)        mode = 2;        // small -> argmax q
        st->vmax = vmax; st->vmin = vmin; st->size = size; st->meanv = (float)mean;
        st->emin = eta_min; st->emax = eta_max;
        st->eta  = 0.5f * (eta_min + eta_max);
        st->eta_star = st->eta;
        st->mode = mode; st->done = (mode != 0) ? 1 : 0; st->iter = 0;
    }
}

// ---------------- f(eta) pass: (T1,T2) at emin, emax, mid ----------------
__global__ void dvp_feval_partial(const f32x4* __restrict__ v4,
                                  float* __restrict__ part,
                                  const DvpState* __restrict__ st, int n4) {
    if (st->done) return;                 // uniform across grid
    const float ea = st->emin, eb = st->emax, ec = st->eta;

    __shared__ f32x4 buf[2][NT];
    __shared__ float sm[NT];
    const int tid = threadIdx.x;
    const long long base = (long long)blockIdx.x * (NT * NCHUNK) + tid;

    float t1a = 0.f, t2a = 0.f, t1b = 0.f, t2b = 0.f, t1c = 0.f, t2c = 0.f;

    if (base < n4) asyncCopy16(v4 + base, &buf[0][tid]);
    COMPILER_BARRIER();
    for (int c = 0; c < NCHUNK; ++c) {
        long long idx = base + (long long)c * NT;
        if (c + 1 < NCHUNK) {
            long long nidx = base + (long long)(c + 1) * NT;
            if (nidx < n4) asyncCopy16(v4 + nidx, &buf[(c + 1) & 1][tid]);
            COMPILER_BARRIER();
            waitAsync1();
        } else {
            waitAsync0();
        }
        f32x4 x = buf[c & 1][tid];
        if (idx < n4) {
            #pragma unroll
            for (int k = 0; k < 4; ++k) {
                float val = (k == 0) ? x.x : (k == 1) ? x.y : (k == 2) ? x.z : x.w;
                float d;
                d = fmaxf(val - ea, 0.f); t1a += d; t2a = fmaf(d, d, t2a);
                d = fmaxf(val - eb, 0.f); t1b += d; t2b = fmaf(d, d, t2b);
                d = fmaxf(val - ec, 0.f); t1c += d; t2c = fmaf(d, d, t2c);
            }
        }
    }
    float r0 = blockSumF(t1a, sm);
    float r1 = blockSumF(t2a, sm);
    float r2 = blockSumF(t1b, sm);
    float r3 = blockSumF(t2b, sm);
    float r4 = blockSumF(t1c, sm);
    float r5 = blockSumF(t2c, sm);
    if (tid == 0) {
        float* p = part + (size_t)blockIdx.x * 8;
        p[0] = r0; p[1] = r1; p[2] = r2; p[3] = r3; p[4] = r4; p[5] = r5;
    }
}

// ---------------- control: expansion / bisection state machine ----------------
__global__ void dvp_update(const float* __restrict__ part,
                           DvpState* __restrict__ st, int nb, double dm) {
    if (st->done) return;
    __shared__ double dsm[NT];
    const int t = threadIdx.x;
    double a1 = 0, a2 = 0, b1 = 0, b2 = 0, c1 = 0, c2 = 0;
    for (int i = t; i < nb; i += NT) {
        const float* p = part + (size_t)i * 8;
        a1 += (double)p[0]; a2 += (double)p[1];
        b1 += (double)p[2]; b2 += (double)p[3];
        c1 += (double)p[4]; c2 += (double)p[5];
    }
    a1 = blockSumD(a1, dsm); a2 = blockSumD(a2, dsm);
    b1 = blockSumD(b1, dsm); b2 = blockSumD(b2, dsm);
    c1 = blockSumD(c1, dsm); c2 = blockSumD(c2, dsm);
    if (t == 0) {
        const double size = (double)st->size;
        // f(eta) = m * sum(p^2) / sum(p)^2 - 1 - size
        double lo = dm * a2 / (a1 * a1) - 1.0 - size;
        double up = dm * b2 / (b1 * b1) - 1.0 - size;
        double vm = dm * c2 / (c1 * c1) - 1.0 - size;
        float emin = st->emin, emax = st->emax, eta = st->eta;
        if (lo > 0.0 || up < 0.0) {
            // expansion body (mirrors reference exp_body exactly)
            float len   = emax - emin;
            float emax1 = (lo > 0.0) ? emin : emax;
            float emin1 = (lo > 0.0) ? (emin - 2.0f * len) : emin;
            float emin2 = (up < 0.0) ? emax1 : emin1;
            float emax2 = (up < 0.0) ? (emax1 + 2.0f * len) : emax1;
            emin = emin2; emax = emax2;
            st->emin = emin; st->emax = emax;
            st->eta  = 0.5f * (emin + emax);
        } else {
            // one bisection step using f(mid)
            st->eta_star = eta;
            if (fabs(vm) <= DVP_TOL) {
                st->done = 1;
            } else {
                if (vm > 0.0)      emax = eta;
                else if (vm < 0.0) emin = eta;
                st->emin = emin; st->emax = emax;
                st->eta  = 0.5f * (emin + emax);
            }
        }
        st->iter = st->iter + 1;
    }
}

// ---------------- final pass: W = sum relu(v-eta*), WV = sum relu*v ----------------
__global__ void dvp_final_partial(const f32x4* __restrict__ v4,
                                  float* __restrict__ part,
                                  const DvpState* __restrict__ st, int n4) {
    if (st->mode != 0) return;
    const float eta = st->eta_star;

    __shared__ f32x4 buf[2][NT];
    __shared__ float sm[NT];
    const int tid = threadIdx.x;
    const long long base = (long long)blockIdx.x * (NT * NCHUNK) + tid;

    float w = 0.f, wv = 0.f;

    if (base < n4) asyncCopy16(v4 + base, &buf[0][tid]);
    COMPILER_BARRIER();
    for (int c = 0; c < NCHUNK; ++c) {
        long long idx = base + (long long)c * NT;
        if (c + 1 < NCHUNK) {
            long long nidx = base + (long long)(c + 1) * NT;
            if (nidx < n4) asyncCopy16(v4 + nidx, &buf[(c + 1) & 1][tid]);
            COMPILER_BARRIER();
            waitAsync1();
        } else {
            waitAsync0();
        }
        f32x4 x = buf[c & 1][tid];
        if (idx < n4) {
            #pragma unroll
            for (int k = 0; k < 4; ++k) {
                float val = (k == 0) ? x.x : (k == 1) ? x.y : (k == 2) ? x.z : x.w;
                float r = fmaxf(val - eta, 0.f);
                w += r;
                wv = fmaf(r, val, wv);
            }
        }
    }
    float bw  = blockSumF(w,  sm);
    float bwv = blockSumF(wv, sm);
    if (tid == 0) {
        float* p = part + (size_t)blockIdx.x * 8;
        p[0] = bw; p[1] = bwv;
    }
}

__global__ void dvp_out(const float* __restrict__ part,
                        const DvpState* __restrict__ st, int nb,
                        float* __restrict__ out) {
    __shared__ double dsm[NT];
    const int t = threadIdx.x;
    const int mode = st->mode;      // uniform
    double w = 0.0, wv = 0.0;
    if (mode == 0) {
        for (int i = t; i < nb; i += NT) {
            const float* p = part + (size_t)i * 8;
            w += (double)p[0]; wv += (double)p[1];
        }
        w  = blockSumD(w,  dsm);
        wv = blockSumD(wv, dsm);
    }
    if (t == 0) {
        float r;
        if (mode == 1)      r = st->meanv;          // uniform q -> mean(v)
        else if (mode == 2) r = st->vmax;           // argmax q  -> vmax
        else                r = (float)(wv / w);    // dot(q, v)
        out[0] = r;
    }
}

// ---------------- host-side launch ----------------
extern "C" void kernel_launch(void* const* d_in, const int* in_sizes, int n_in,
                              void* d_out, int out_size, void* d_ws, size_t ws_size,
                              hipStream_t stream) {
    const float* v = (const float*)d_in[0];
    const int n  = in_sizes[0];
    const int n4 = n >> 2;                       // M = 2^24 -> divisible by 4
    const int nb = (n4 + NT * NCHUNK - 1) / (NT * NCHUNK);
    const double dm = (double)n;

    DvpState* st   = (DvpState*)d_ws;
    float*    part = (float*)((char*)d_ws + 256);

    dvp_stats_partial<<<nb, NT, 0, stream>>>((const f32x4*)v, part, n4);
    dvp_stats_final<<<1, NT, 0, stream>>>(part, st, nb, dm);

    for (int i = 0; i < N_ITER; ++i) {
        dvp_feval_partial<<<nb, NT, 0, stream>>>((const f32x4*)v, part, st, n4);
        dvp_update<<<1, NT, 0, stream>>>(part, st, nb, dm);
    }

    dvp_final_partial<<<nb, NT, 0, stream>>>((const f32x4*)v, part, st, n4);
    dvp_out<<<1, NT, 0, stream>>>(part, st, nb, (float*)d_out);
}